// CPCircuitLayer_66168266162577
// MI455X (gfx1250) — compile-verified
//
#include <hip/hip_runtime.h>
#include <hip/hip_bf16.h>

// Problem constants (reference: B=2, S=1024, H=1024, R=32)
#define Bdim 2
#define Sdim 1024
#define Hdim 1024
#define Rdim 32

typedef __attribute__((ext_vector_type(2))) float v2f;
typedef __attribute__((ext_vector_type(8))) float v8f;

// ---------------------------------------------------------------------------
// Phase 1a: A[b*S+s, r] = sum_h hidden[(b*S+s)*H + h] * W_seq[r*H + h]
// M = B*S = 2048, N = R = 32, K = H = 1024.
// One wave (32 lanes) computes one 16x16 tile via V_WMMA_F32_16X16X4_F32.
//
// f32 WMMA fragment layout (ISA 7.12.2):
//   A (16x4):  lane&15 = M row; VGPR0 holds K=0 (lanes 0-15) / K=2 (lanes 16-31),
//              VGPR1 holds K=1 / K=3  -> per-lane float2 at col k + (lane>>4)*2
//   B (4x16):  lane&15 = N col; same K striping -> identical float2 pattern on
//              W_seq rows (W_seq is [R,H] row-major, B[k,n] = W_seq[n,k])
//   C/D 16x16: VGPR g holds M = g + 8*(lane>>4), N = lane&15
// ---------------------------------------------------------------------------
__global__ __launch_bounds__(32) void cp_gemm_A(const float* __restrict__ hidden,
                                                const float* __restrict__ Wseq,
                                                float* __restrict__ Aout) {
    const int lane = threadIdx.x;            // wave32
    const int tile = blockIdx.x;             // 256 tiles: 128 M-tiles x 2 N-tiles
    const int n0   = (tile & 1) * 16;
    const int m0   = (tile >> 1) * 16;
    const int lm   = lane & 15;
    const int koff = (lane >> 4) * 2;        // K-pair select: 0 or 2

    const float* arow = hidden + (size_t)(m0 + lm) * Hdim + koff;
    const float* brow = Wseq   + (size_t)(n0 + lm) * Hdim + koff;

    v8f acc = {};
    for (int k = 0; k < Hdim; k += 16) {
#pragma unroll
        for (int kk = 0; kk < 16; kk += 4) {
            v2f a = *(const v2f*)(arow + k + kk);   // global_load_b64
            v2f b = *(const v2f*)(brow + k + kk);   // global_load_b64
            acc = __builtin_amdgcn_wmma_f32_16x16x4_f32(
                false, a, false, b, (short)0, acc, false, false);
        }
    }

#pragma unroll
    for (int g = 0; g < 8; ++g) {
        const int m = m0 + g + ((lane >> 4) << 3);
        Aout[(size_t)m * Rdim + n0 + lm] = acc[g];
    }
}

// ---------------------------------------------------------------------------
// Phase 1b: Bm[b, h, r] = sum_s hidden[b, s, h] * W_hid[r*S + s]
// Per batch: M = H (h), N = R, K = S (s). A-matrix is hidden[b] transposed:
// element [m=h, k=s] sits at hidden_b[s*H + h] (stride H along K -> two b32
// loads per fragment). B-matrix [k=s, n=r] = W_hid[n*S + k] (contiguous).
// ---------------------------------------------------------------------------
__global__ __launch_bounds__(32) void cp_gemm_B(const float* __restrict__ hidden,
                                                const float* __restrict__ Whid,
                                                float* __restrict__ Bout) {
    const int lane = threadIdx.x;
    const int tile = blockIdx.x;             // 256 tiles: b(2) x 64 M-tiles x 2 N-tiles
    const int n0   = (tile & 1) * 16;
    const int m0   = ((tile >> 1) & 63) * 16;
    const int b    = tile >> 7;
    const int lm   = lane & 15;
    const int koff = (lane >> 4) * 2;

    const float* hb   = hidden + (size_t)b * Sdim * Hdim;
    const float* acol = hb + (m0 + lm);                       // + k*Hdim per step
    const float* brow = Whid + (size_t)(n0 + lm) * Sdim + koff;

    v8f acc = {};
    for (int k = 0; k < Sdim; k += 16) {
#pragma unroll
        for (int kk = 0; kk < 16; kk += 4) {
            const int kb = k + kk + koff;
            v2f a;
            a[0] = acol[(size_t)(kb + 0) * Hdim];             // global_load_b32 (strided)
            a[1] = acol[(size_t)(kb + 1) * Hdim];
            v2f bf = *(const v2f*)(brow + k + kk);            // global_load_b64
            acc = __builtin_amdgcn_wmma_f32_16x16x4_f32(
                false, a, false, bf, (short)0, acc, false, false);
        }
    }

#pragma unroll
    for (int g = 0; g < 8; ++g) {
        const int m = m0 + g + ((lane >> 4) << 3);
        Bout[((size_t)b * Hdim + m) * Rdim + n0 + lm] = acc[g];
    }
}

// ---------------------------------------------------------------------------
// Phase 2: out[b, i] = dot32(A[b, idx_s[i], :], Bm[b, idx_h[i], :])
// One thread per index pair i; index (int64 pair, 16B) loaded once with a
// single b128 load and reused for both batches. Gather tables A/Bm total
// 512 KB -> resident in the 192 MB L2, so the 512 B/thread of table reads are
// L2 hits; HBM traffic is just indices (16 MB) + output (8 MB).
// ---------------------------------------------------------------------------
__global__ __launch_bounds__(256) void cp_gather_dot(const long long* __restrict__ idx,
                                                     const float* __restrict__ A,
                                                     const float* __restrict__ Bm,
                                                     float* __restrict__ out) {
    const size_t i = (size_t)blockIdx.x * 256 + threadIdx.x;  // 0 .. S*H-1

    // Prefetch the index stream ~32KB ahead (global_prefetch_b8).
    __builtin_prefetch(idx + 2 * i + 4096, 0, 1);

    const longlong2 ip = *(const longlong2*)(idx + 2 * i);    // global_load_b128
    const int is = (int)ip.x;   // row into A (per batch)
    const int ih = (int)ip.y;   // row into Bm (per batch)

    const float4* ar0 = (const float4*)(A  + (size_t)is * Rdim);
    const float4* ar1 = (const float4*)(A  + (size_t)(Sdim + is) * Rdim);
    const float4* br0 = (const float4*)(Bm + (size_t)ih * Rdim);
    const float4* br1 = (const float4*)(Bm + (size_t)(Hdim + ih) * Rdim);

    float acc0 = 0.0f, acc1 = 0.0f;
#pragma unroll
    for (int q = 0; q < Rdim / 4; ++q) {
        const float4 a0 = ar0[q], b0 = br0[q];                // global_load_b128 x2
        acc0 = fmaf(a0.x, b0.x, acc0);
        acc0 = fmaf(a0.y, b0.y, acc0);
        acc0 = fmaf(a0.z, b0.z, acc0);
        acc0 = fmaf(a0.w, b0.w, acc0);
        const float4 a1 = ar1[q], b1 = br1[q];
        acc1 = fmaf(a1.x, b1.x, acc1);
        acc1 = fmaf(a1.y, b1.y, acc1);
        acc1 = fmaf(a1.z, b1.z, acc1);
        acc1 = fmaf(a1.w, b1.w, acc1);
    }

    out[i]                            = acc0;   // b = 0
    out[(size_t)Sdim * Hdim + i]      = acc1;   // b = 1
}

// ---------------------------------------------------------------------------
extern "C" void kernel_launch(void* const* d_in, const int* in_sizes, int n_in,
                              void* d_out, int out_size, void* d_ws, size_t ws_size,
                              hipStream_t stream) {
    const float*     hidden = (const float*)d_in[0];      // (B,S,H) f32
    const float*     Wseq   = (const float*)d_in[1];      // (R,H)   f32
    const float*     Whid   = (const float*)d_in[2];      // (R,S)   f32
    const long long* idx    = (const long long*)d_in[3];  // (S*H,2) int64
    float*           out    = (float*)d_out;              // (B,S,H) f32

    // Workspace: A (B*S*R f32 = 256KB) then Bm (B*H*R f32 = 256KB).
    float* Aws = (float*)d_ws;
    float* Bws = Aws + (size_t)Bdim * Sdim * Rdim;

    // Phase 1: WMMA GEMMs (independent; trivial cost vs. phase 2).
    cp_gemm_A<<<(Bdim * Sdim / 16) * (Rdim / 16), 32, 0, stream>>>(hidden, Wseq, Aws);
    cp_gemm_B<<<Bdim * (Hdim / 16) * (Rdim / 16), 32, 0, stream>>>(hidden, Whid, Bws);

    // Phase 2: gather + dot, one thread per index pair, both batches per thread.
    cp_gather_dot<<<(Sdim * Hdim) / 256, 256, 0, stream>>>(idx, Aws, Bws, out);
}